// GraphEncoder_24326694765010
// MI455X (gfx1250) — compile-verified
//
#include <hip/hip_runtime.h>

typedef __attribute__((ext_vector_type(2))) float v2f;
typedef __attribute__((ext_vector_type(8))) float v8f;

#define DD 128  // feature dim (in_dim == hidden_dim)

// ---------------- utility kernels ----------------

__global__ void k_fill(float* __restrict__ p, float v, int n) {
    int i = blockIdx.x * blockDim.x + threadIdx.x;
    if (i < n) p[i] = v;
}

// deg[dst] += 1 for every directed edge (deg pre-initialized to 1.0 for self-loops)
__global__ void k_deg(const int* __restrict__ dst, float* __restrict__ deg, int E) {
    int e = blockIdx.x * blockDim.x + threadIdx.x;
    if (e < E) atomicAdd(&deg[dst[e]], 1.0f);
}

// dinv = rsqrt(deg)   (deg >= 1 always because of self-loops)
__global__ void k_rsqrt(float* __restrict__ d, int n) {
    int i = blockIdx.x * blockDim.x + threadIdx.x;
    if (i < n) d[i] = rsqrtf(d[i]);
}

// ---------------- fp32 WMMA GEMM: C[N x 128] = A[N x 128] @ W[128 x 128] ----------------
// One block = 16-row strip; 8 waves, wave w computes the 16x16 tile at columns [16w, 16w+16).
// A strip staged through LDS. K loop steps by 4 using V_WMMA_F32_16X16X4_F32.
__global__ __launch_bounds__(256) void k_gemm_wmma(const float* __restrict__ A,
                                                   const float* __restrict__ W,
                                                   float* __restrict__ C,
                                                   int nrows) {
    __shared__ float sA[16 * DD];              // 8 KB A strip
    const int tid  = threadIdx.x;
    const int wave = tid >> 5;                 // 0..7 -> column tile
    const int lane = tid & 31;
    const int row0 = blockIdx.x * 16;

    // cooperative load of 16x128 A strip (512 float4s, 2 per thread)
    for (int i = tid; i < 16 * DD / 4; i += 256) {
        int r  = i >> 5;                       // row within strip
        int c4 = i & 31;                       // float4 column
        float4 v = make_float4(0.f, 0.f, 0.f, 0.f);
        int gr = row0 + r;
        if (gr < nrows) v = ((const float4*)A)[(size_t)gr * (DD / 4) + c4];
        ((float4*)sA)[i] = v;
    }
    __syncthreads();

    const int m = lane & 15;                   // A row within tile / B,C column
    const int g = lane >> 4;                   // lane group (K split / row split)
    const int ncol = wave * 16 + m;            // global output column

    v8f acc = {};
    for (int k0 = 0; k0 < DD; k0 += 4) {
        v2f a, b;
        // A 16x4 fp32 layout: lanes 0-15 hold {K=0,K=1}, lanes 16-31 hold {K=2,K=3}
        a.x = sA[m * DD + k0 + 2 * g];
        a.y = sA[m * DD + k0 + 2 * g + 1];
        // B 4x16 fp32 layout: lane holds column n=m, K pair {2g, 2g+1}
        b.x = W[(size_t)(k0 + 2 * g) * DD + ncol];
        b.y = W[(size_t)(k0 + 2 * g + 1) * DD + ncol];
        acc = __builtin_amdgcn_wmma_f32_16x16x4_f32(
            /*neg_a=*/false, a, /*neg_b=*/false, b,
            /*c_mod=*/(short)0, acc, /*reuse_a=*/false, /*reuse_b=*/false);
    }

    // C/D 16x16 fp32 layout: VGPR v holds row (v + 8g), column n
    for (int v = 0; v < 8; ++v) {
        int r = row0 + v + 8 * g;
        if (r < nrows) C[(size_t)r * DD + ncol] = acc[v];
    }
}

// ---------------- edge scatter: AGG[dst] += H[src] * dinv[src]*dinv[dst] ----------------
// One wave per edge: uniform src/dst (scalar), lanes gather float4 (coalesced 512B)
// and issue 4 global fp32 atomic adds each.
__global__ __launch_bounds__(256) void k_edge(const float* __restrict__ H,
                                              float* __restrict__ AGG,
                                              const float* __restrict__ dinv,
                                              const int* __restrict__ src,
                                              const int* __restrict__ dst,
                                              int E) {
    const int e = blockIdx.x * (blockDim.x >> 5) + (threadIdx.x >> 5);
    const int lane = threadIdx.x & 31;
    if (e >= E) return;
    const int s = src[e];
    const int d = dst[e];
    const float w = dinv[s] * dinv[d];
    float4 v = ((const float4*)(H + (size_t)s * DD))[lane];
    float* ad = AGG + (size_t)d * DD + lane * 4;
    atomicAdd(ad + 0, v.x * w);
    atomicAdd(ad + 1, v.y * w);
    atomicAdd(ad + 2, v.z * w);
    atomicAdd(ad + 3, v.w * w);
}

// ---------------- self-loop + bias + ReLU (in place on AGG) ----------------
// AGG[i,j] = relu(AGG[i,j] + H[i,j]*dinv[i]^2 + b[j])
__global__ void k_post(const float* __restrict__ H, float* __restrict__ A,
                       const float* __restrict__ dinv, const float* __restrict__ bias,
                       int n) {
    int i = blockIdx.x * blockDim.x + threadIdx.x;
    if (i < n) {
        int row = i >> 7;
        int col = i & (DD - 1);
        float di = dinv[row];
        float v = A[i] + H[i] * di * di + bias[col];
        A[i] = v > 0.f ? v : 0.f;
    }
}

// ---------------- global mean pool ----------------
__global__ void k_pool(const float* __restrict__ h, const int* __restrict__ batch,
                       float* __restrict__ psum, float* __restrict__ pcnt, int n) {
    int i = blockIdx.x * blockDim.x + threadIdx.x;
    if (i < n) {
        int row = i >> 7;
        int col = i & (DD - 1);
        int b = batch[row];
        atomicAdd(&psum[b * DD + col], h[i]);
        if (col == 0) atomicAdd(&pcnt[b], 1.0f);
    }
}

__global__ void k_pool_div(const float* __restrict__ psum, const float* __restrict__ pcnt,
                           float* __restrict__ out, int n) {
    int i = blockIdx.x * blockDim.x + threadIdx.x;
    if (i < n) {
        int b = i >> 7;
        out[i] = psum[i] / fmaxf(pcnt[b], 1.0f);
    }
}

// ---------------- driver ----------------

extern "C" void kernel_launch(void* const* d_in, const int* in_sizes, int n_in,
                              void* d_out, int out_size, void* d_ws, size_t ws_size,
                              hipStream_t stream) {
    const float* x   = (const float*)d_in[0];   // [N, 128]
    const float* W1  = (const float*)d_in[1];   // [128, 128]
    const float* b1  = (const float*)d_in[2];   // [128]
    const float* W2  = (const float*)d_in[3];   // [128, 128]
    const float* b2  = (const float*)d_in[4];   // [128]
    const int*  eidx = (const int*)d_in[5];     // [2, E]
    const int*  batch= (const int*)d_in[6];     // [N]

    const int E = in_sizes[5] / 2;
    const int N = in_sizes[6];
    const int B = out_size / DD;
    const int* src = eidx;
    const int* dst = eidx + E;

    float* ws   = (float*)d_ws;
    float* dinv = ws;                            // N
    float* bufH = dinv + N;                      // N*128 (GEMM output)
    float* bufA = bufH + (size_t)N * DD;         // N*128 (aggregation / layer output)
    float* psum = bufA + (size_t)N * DD;         // B*128
    float* pcnt = psum + (size_t)B * DD;         // B

    const int T = 256;
    const int nND       = N * DD;
    const int gridND    = (nND + T - 1) / T;
    const int gridN     = (N + T - 1) / T;
    const int gridE     = (E + T - 1) / T;
    const int gridEdge  = (E + 7) / 8;           // 8 edges (waves) per block
    const int gridGemm  = (N + 15) / 16;

    // --- degree / symmetric norm ---
    k_fill<<<gridN, T, 0, stream>>>(dinv, 1.0f, N);                 // self-loop contributes 1
    k_deg<<<gridE, T, 0, stream>>>(dst, dinv, E);
    k_rsqrt<<<gridN, T, 0, stream>>>(dinv, N);

    // --- layer 1 ---
    k_gemm_wmma<<<gridGemm, T, 0, stream>>>(x, W1, bufH, N);
    k_fill<<<gridND, T, 0, stream>>>(bufA, 0.0f, nND);
    k_edge<<<gridEdge, T, 0, stream>>>(bufH, bufA, dinv, src, dst, E);
    k_post<<<gridND, T, 0, stream>>>(bufH, bufA, dinv, b1, nND);    // bufA = h1

    // --- layer 2 ---
    k_gemm_wmma<<<gridGemm, T, 0, stream>>>(bufA, W2, bufH, N);
    k_fill<<<gridND, T, 0, stream>>>(bufA, 0.0f, nND);
    k_edge<<<gridEdge, T, 0, stream>>>(bufH, bufA, dinv, src, dst, E);
    k_post<<<gridND, T, 0, stream>>>(bufH, bufA, dinv, b2, nND);    // bufA = h2

    // --- global mean pool ---
    k_fill<<<(B * DD + T - 1) / T, T, 0, stream>>>(psum, 0.0f, B * DD);
    k_fill<<<(B + T - 1) / T, T, 0, stream>>>(pcnt, 0.0f, B);
    k_pool<<<gridND, T, 0, stream>>>(bufA, batch, psum, pcnt, nND);
    k_pool_div<<<(B * DD + T - 1) / T, T, 0, stream>>>(psum, pcnt, (float*)d_out, B * DD);
}